// LocallyConnected1D_78176994721888
// MI455X (gfx1250) — compile-verified
//
#include <hip/hip_runtime.h>

// LocallyConnected1D: out[b,l,n] = sum_k patches[b,l,k] * W[l,k,n] + bias[l,n]
// B=32, L=512, C=256, K_LEN=3 -> KC=768, NB_FILTER=256, OUT_LEN=510. All fp32.
//
// Strategy (MI455X / gfx1250, wave32):
//  - HBM-bound on W (402 MB, read exactly once device-wide) -> stream W with
//    non-temporal loads so it doesn't churn the 192MB L2; out stores also NT.
//  - Per (l, N-half) block: stage 32x768 patch tile in LDS (padded stride 772
//    dwords -> conflict-free ds_load_2addr_b64 A-fragments), accumulate with
//    v_wmma_f32_16x16x4_f32 (full fp32; compute ceiling ~= 19us memory floor,
//    so no precision downgrade needed).

typedef __attribute__((ext_vector_type(2))) float v2f;
typedef __attribute__((ext_vector_type(4))) float v4f;
typedef __attribute__((ext_vector_type(8))) float v8f;

namespace {
constexpr int kB       = 32;
constexpr int kL       = 512;
constexpr int kC       = 256;
constexpr int kKLen    = 3;
constexpr int kNB      = 256;
constexpr int kOutLen  = kL - kKLen + 1;   // 510
constexpr int kKC      = kKLen * kC;       // 768
constexpr int kLdsStr  = kKC + 4;          // 772 dwords: lane r -> banks {4r..4r+3}
constexpr int kNSplit  = 2;                // blocks per l along N
constexpr int kNPerBlk = kNB / kNSplit;    // 128 -> 8 waves x 16 columns
}  // namespace

__global__ __launch_bounds__(256)
void lc1d_wmma_f32(const float* __restrict__ x,
                   const float* __restrict__ W,
                   const float* __restrict__ bias,
                   float* __restrict__ out) {
  extern __shared__ float lds[];  // kB * kLdsStr floats = 98816 bytes

  const int l   = blockIdx.x;   // output position
  const int ny  = blockIdx.y;   // N half
  const int tid = threadIdx.x;

  // ---- Stage patches: row b = 768 contiguous floats at x[(b*L + l)*C] ----
  // 32 rows * 192 float4 chunks; padded row base (row*772*4 B) is 16B aligned.
  for (int c = tid; c < kB * (kKC / 4); c += 256) {
    const int row = c / (kKC / 4);
    const int j   = (c % (kKC / 4)) * 4;
    v4f v = *(const v4f*)(x + (size_t)(row * kL + l) * kC + j);
    *(v4f*)(lds + row * kLdsStr + j) = v;
  }
  __syncthreads();

  const int lane   = tid & 31;
  const int wave   = tid >> 5;
  const int hi     = lane >> 4;        // 0: lanes 0-15, 1: lanes 16-31
  const int lane16 = lane & 15;
  const int ncol   = ny * kNPerBlk + wave * 16 + lane16;  // this lane's N column

  const float* __restrict__ Wl = W + (size_t)l * kKC * kNB;  // [768][256]

  // A-fragment base addresses (per ISA 7.12.2: M = lane%16, K split by lane half)
  const float* ldsA0 = lds + lane16 * kLdsStr;          // batches 0..15
  const float* ldsA1 = lds + (lane16 + 16) * kLdsStr;   // batches 16..31

  v8f acc0 = {};  // D for M-tile 0 (batch 0..15)
  v8f acc1 = {};  // D for M-tile 1 (batch 16..31)

#pragma unroll 4
  for (int k0 = 0; k0 < kKC; k0 += 4) {
    const int kk = k0 + hi * 2;  // lanes 0-15 read K={k0,k0+1}, lanes 16-31 K={k0+2,k0+3}
    v2f a0 = *(const v2f*)(ldsA0 + kk);   // ds_load (2addr_b64), conflict-free
    v2f a1 = *(const v2f*)(ldsA1 + kk);
    // B-fragment: VGPR0 = row kk, VGPR1 = row kk+1, N = lane%16 (mirrors A's K split)
    // W is streamed exactly once -> non-temporal loads keep it out of L2.
    const float* wp = Wl + (size_t)kk * kNB + ncol;
    v2f b;
    b.x = __builtin_nontemporal_load(wp);
    b.y = __builtin_nontemporal_load(wp + kNB);
    acc0 = __builtin_amdgcn_wmma_f32_16x16x4_f32(
        false, a0, false, b, (short)0, acc0, false, false);
    acc1 = __builtin_amdgcn_wmma_f32_16x16x4_f32(
        false, a1, false, b, (short)0, acc1, false, false);
  }

  // ---- Bias + store. C/D layout: VGPR r -> M = r (+8 for lanes 16-31), N = lane%16.
  const float bv = bias[(size_t)l * kNB + ncol];
#pragma unroll
  for (int r = 0; r < 8; ++r) {
    const int m0 = r + hi * 8;        // batch index for acc0 element r
    const int m1 = m0 + 16;           // batch index for acc1 element r
    // Output is written once and never re-read -> non-temporal stores.
    __builtin_nontemporal_store(acc0[r] + bv,
        out + ((size_t)m0 * kOutLen + l) * kNB + ncol);
    __builtin_nontemporal_store(acc1[r] + bv,
        out + ((size_t)m1 * kOutLen + l) * kNB + ncol);
  }
}

extern "C" void kernel_launch(void* const* d_in, const int* in_sizes, int n_in,
                              void* d_out, int out_size, void* d_ws, size_t ws_size,
                              hipStream_t stream) {
  (void)in_sizes; (void)n_in; (void)out_size; (void)d_ws; (void)ws_size;
  const float* x  = (const float*)d_in[0];
  const float* W  = (const float*)d_in[1];
  const float* b  = (const float*)d_in[2];
  float* out      = (float*)d_out;

  dim3 grid(kOutLen, kNSplit, 1);
  const size_t shmem = (size_t)kB * kLdsStr * sizeof(float);  // 98816 B (< 320 KB/WGP)
  lc1d_wmma_f32<<<grid, 256, shmem, stream>>>(x, W, b, out);
}